// KAttentionNetwork_17746804867962
// MI455X (gfx1250) — compile-verified
//
#include <hip/hip_runtime.h>
#include <hip/hip_bf16.h>

// Problem constants (fixed by the reference)
#define NN   196608
#define BB   4096
#define HH   230
#define HP   232          // padded LDS row stride
#define CC   53
#define KP   464          // 2*H padded to multiple of 4 for WMMA K-steps
#define MAXBAG 64

// Workspace layout (float offsets)
#define OFF_CLS_ADD 0        // [2][C] per-class logit additive term (incl. b_s)
#define OFF_CLS_VSM 128      // [2][C] per-class 2-way softmax of att weights
#define OFF_DISC_T  256      // [464][64] zero-padded transposed disc matrix
#define OFF_TOWER   30720    // [B][464] tower rows (zero-padded K)

typedef __attribute__((ext_vector_type(2))) float v2f;
typedef __attribute__((ext_vector_type(8))) float v8f;

// ---------------------------------------------------------------------------
// Kernel A: per-class precompute.
// contrib[k][c] = ws[k,H:2H]·tanh(rel0[l0]) + ws[k,2H:3H]·tanh(rel1[l1]) + b_s[k]
// vsm[.][c]    = softmax over k of (att_w0[l0], att_w1[l1])
// ---------------------------------------------------------------------------
__global__ __launch_bounds__(256) void cls_prep_kernel(
    const float* __restrict__ rel0, const float* __restrict__ rel1,
    const float* __restrict__ aw0,  const float* __restrict__ aw1,
    const float* __restrict__ w_s,  const float* __restrict__ b_s,
    const int*   __restrict__ rlv,  float* __restrict__ ws)
{
    __shared__ float rA[256];
    __shared__ float rB[256];
    const int c   = blockIdx.x;
    const int tid = threadIdx.x;
    const int l0 = rlv[c * 2 + 0];
    const int l1 = rlv[c * 2 + 1];
    float a = 0.f, b = 0.f;
    if (tid < HH) {
        float t0 = tanhf(rel0[l0 * HH + tid]);
        float t1 = tanhf(rel1[l1 * HH + tid]);
        // w_s rows have stride 3H = 690
        a = t0 * w_s[HH + tid]       + t1 * w_s[2 * HH + tid];
        b = t0 * w_s[690 + HH + tid] + t1 * w_s[690 + 2 * HH + tid];
    }
    rA[tid] = a; rB[tid] = b;
    __syncthreads();
    for (int s = 128; s > 0; s >>= 1) {
        if (tid < s) { rA[tid] += rA[tid + s]; rB[tid] += rB[tid + s]; }
        __syncthreads();
    }
    if (tid == 0) {
        ws[OFF_CLS_ADD + c]      = rA[0] + b_s[0];
        ws[OFF_CLS_ADD + CC + c] = rB[0] + b_s[1];
        float v0 = aw0[l0], v1 = aw1[l1];
        float m  = fmaxf(v0, v1);
        float e0 = expf(v0 - m), e1 = expf(v1 - m);
        float inv = 1.f / (e0 + e1);
        ws[OFF_CLS_VSM + c]      = e0 * inv;
        ws[OFF_CLS_VSM + CC + c] = e1 * inv;
    }
}

// ---------------------------------------------------------------------------
// Kernel D: pad + transpose disc_matrix (C x 2H, row-major) -> disc_t (464 x 64)
// disc_t[k][n] = disc[n][k] (zero outside valid range)
// ---------------------------------------------------------------------------
__global__ __launch_bounds__(256) void disc_prep_kernel(
    const float* __restrict__ disc, float* __restrict__ disc_t)
{
    int idx = blockIdx.x * 256 + threadIdx.x;
    if (idx >= KP * 64) return;
    int k = idx >> 6;
    int n = idx & 63;
    float v = 0.f;
    if (k < 2 * HH && n < CC) v = disc[n * (2 * HH) + k];
    disc_t[idx] = v;
}

// ---------------------------------------------------------------------------
// Kernel B: fused per-bag kernel. One block per bag; x rows staged to LDS and
// read exactly once from HBM. Produces tower row (464 floats, zero-padded).
// ---------------------------------------------------------------------------
__global__ __launch_bounds__(256) void bag_kernel(
    const float* __restrict__ x, const int* __restrict__ label,
    const int* __restrict__ scope, const float* __restrict__ w_s,
    const float* __restrict__ ws, float* __restrict__ tower)
{
    __shared__ float xs[MAXBAG][HP];
    __shared__ float wsl[2][HP];
    __shared__ float lg[2][MAXBAG];
    __shared__ float va[2][MAXBAG];
    __shared__ float sinv[2];
    __shared__ float lat[2];

    const int b     = blockIdx.x;
    const int tid   = threadIdx.x;
    const int wave  = tid >> 5;
    const int lane  = tid & 31;
    const int start = scope[b];
    int count = scope[b + 1] - start;
    if (count > MAXBAG) count = MAXBAG;

    // stage w_s first-H rows
    if (tid < HH) {
        wsl[0][tid] = w_s[tid];
        wsl[1][tid] = w_s[690 + tid];
    }
    __syncthreads();

    // Phase 1: each wave handles instances j = wave, wave+8, ...
    for (int j = wave; j < count; j += 8) {
        const int g = start + j;
        const float* xr = x + (long)g * HH;
        float a0 = 0.f, a1 = 0.f;
        for (int h = lane; h < HH; h += 32) {
            float v = xr[h];
            xs[j][h] = v;
            float t = tanhf(v);
            a0 += t * wsl[0][h];
            a1 += t * wsl[1][h];
        }
        for (int off = 16; off > 0; off >>= 1) {
            a0 += __shfl_xor(a0, off, 32);
            a1 += __shfl_xor(a1, off, 32);
        }
        if (lane == 0) {
            int cls = label[g];
            lg[0][j] = a0 + ws[OFF_CLS_ADD + cls];
            lg[1][j] = a1 + ws[OFF_CLS_ADD + CC + cls];
            va[0][j] = ws[OFF_CLS_VSM + cls];
            va[1][j] = ws[OFF_CLS_VSM + CC + cls];
        }
    }
    __syncthreads();

    // Phase 2: per-k softmax over the bag + layer attention (tiny, 2 threads)
    if (tid < 2) {
        const int k = tid;
        float m = -1e30f;
        for (int j = 0; j < count; ++j) m = fmaxf(m, lg[k][j]);
        float s = 0.f;
        for (int j = 0; j < count; ++j) { float e = expf(lg[k][j] - m); lg[k][j] = e; s += e; }
        float la = 0.f;
        for (int j = 0; j < count; ++j) la += va[k][j];
        sinv[k] = 1.f / s;
        lat[k]  = la / (float)count;
    }
    __syncthreads();

    // Phase 3: rep = sum_j score*x, scale by layer_att, write tower row
    if (tid < HH) {
        const int h = tid;
        float r0 = 0.f, r1 = 0.f;
        for (int j = 0; j < count; ++j) {
            float xv = xs[j][h];
            r0 += lg[0][j] * xv;
            r1 += lg[1][j] * xv;
        }
        tower[(long)b * KP + h]      = lat[0] * sinv[0] * r0;
        tower[(long)b * KP + HH + h] = lat[1] * sinv[1] * r1;
    }
    if (tid < KP - 2 * HH) tower[(long)b * KP + 2 * HH + tid] = 0.f;
}

// ---------------------------------------------------------------------------
// Kernel C: out = tower @ disc_t + bias via V_WMMA_F32_16X16X4_F32.
// One wave per 16x16 output tile; M-tiles=256, N-tiles=4 (N padded 53->64).
// EXEC is all-1s for the entire WMMA loop (store predication only after).
// ---------------------------------------------------------------------------
__global__ __launch_bounds__(256) void gemm_kernel(
    const float* __restrict__ tower, const float* __restrict__ disc_t,
    const float* __restrict__ bias, float* __restrict__ out)
{
    const int wave = (blockIdx.x * 256 + threadIdx.x) >> 5;  // 0..1023
    const int lane = threadIdx.x & 31;
    const int nt   = wave & 3;
    const int mt   = wave >> 2;
    const int half = lane >> 4;   // K-pair select
    const int lq   = lane & 15;

    const float* arow = tower + (long)(mt * 16 + lq) * KP;
    const int colB = nt * 16 + lq;

    v8f acc = {};
    for (int k0 = 0; k0 < KP; k0 += 4) {
        const int ka = k0 + 2 * half;
        v2f a; a.x = arow[ka];                 a.y = arow[ka + 1];
        v2f bf; bf.x = disc_t[ka * 64 + colB]; bf.y = disc_t[(ka + 1) * 64 + colB];
        acc = __builtin_amdgcn_wmma_f32_16x16x4_f32(
            false, a, false, bf, (short)0, acc, false, false);
    }

    if (colB < CC) {
        const float bv = bias[colB];
#pragma unroll
        for (int r = 0; r < 8; ++r) {
            const int row = mt * 16 + r + 8 * half;
            out[(long)row * CC + colB] = acc[r] + bv;
        }
    }
}

// ---------------------------------------------------------------------------
extern "C" void kernel_launch(void* const* d_in, const int* in_sizes, int n_in,
                              void* d_out, int out_size, void* d_ws, size_t ws_size,
                              hipStream_t stream)
{
    const float* x     = (const float*)d_in[0];
    const float* rel0  = (const float*)d_in[1];
    const float* rel1  = (const float*)d_in[2];
    const float* aw0   = (const float*)d_in[3];
    const float* aw1   = (const float*)d_in[4];
    const float* w_s   = (const float*)d_in[5];
    const float* b_s   = (const float*)d_in[6];
    const float* disc  = (const float*)d_in[7];
    const float* bias  = (const float*)d_in[8];
    const int*   label = (const int*)d_in[9];
    const int*   rlv   = (const int*)d_in[10];
    const int*   scope = (const int*)d_in[11];

    float* ws  = (float*)d_ws;
    float* out = (float*)d_out;
    float* disc_t = ws + OFF_DISC_T;
    float* tower  = ws + OFF_TOWER;

    const int numBags = in_sizes[11] - 1;   // 4096

    // A: per-class precompute (53 blocks)
    cls_prep_kernel<<<CC, 256, 0, stream>>>(rel0, rel1, aw0, aw1, w_s, b_s, rlv, ws);
    // D: disc transpose + pad (464*64 elements)
    disc_prep_kernel<<<(KP * 64 + 255) / 256, 256, 0, stream>>>(disc, disc_t);
    // B: fused per-bag kernel — reads x exactly once
    bag_kernel<<<numBags, 256, 0, stream>>>(x, label, scope, w_s, ws, tower);
    // C: WMMA f32 GEMM (4096x464)@(464x64) -> write 53 valid columns
    gemm_kernel<<<(numBags / 16) * 4 / 8, 256, 0, stream>>>(tower, disc_t, bias, out);
}